// TimeDecayLoss_52871047414308
// MI455X (gfx1250) — compile-verified
//
#include <hip/hip_runtime.h>

#define Hh 260
#define Ww 346
#define Bb 16
#define Tt 4096
#define Cc 1212          // 2H + 2W
#define SEGTOT 606       // H + W (channels actually needed per event)
#define BLK 256
#define NWAVE 8
#define TEMP_F 256.0f
#define WIN_THR (30.0f * 256.0f)   // exp(-30) ~ 9e-14, below f32 noise

__device__ __forceinline__ float waveSum(float v) {
#pragma unroll
  for (int o = 16; o > 0; o >>= 1) v += __shfl_xor(v, o, 32);
  return v;
}
__device__ __forceinline__ float waveMax(float v) {
#pragma unroll
  for (int o = 16; o > 0; o >>= 1) v = fmaxf(v, __shfl_xor(v, o, 32));
  return v;
}

__global__ __launch_bounds__(BLK) void tdloss_main(
    const float* __restrict__ pred, const float* __restrict__ target,
    float* __restrict__ outbuf, int atomicMode) {
  __shared__ __align__(16) float sPred[SEGTOT];
  __shared__ __align__(16) float sS[SEGTOT];
  __shared__ float red[3 * NWAVE];

  const int bid = blockIdx.x;           // = b*T + t
  const int t   = bid & (Tt - 1);
  const int b   = bid >> 12;
  const int tid  = threadIdx.x;
  const int lane = tid & 31;
  const int wid  = tid >> 5;

  // this event's record (uniform address -> scalar load)
  const float4 ev = ((const float4*)target)[bid];
  const float t0 = ev.x;
  const int   p  = (int)ev.w;

  for (int i = tid; i < SEGTOT; i += BLK) sS[i] = 0.0f;

  // ---- CDNA5 async global->LDS staging of the two needed pred segments ----
  // segment A: pred channels [p*H, p*H+H)        -> sPred[0..260)
  // segment B: pred channels [2H + p*W, ...+W)   -> sPred[260..606)
  const float* rowBase = pred + (size_t)bid * Cc;
  const float* gA = rowBase + p * Hh;
  const float* gB = rowBase + 2 * Hh + p * Ww;
  for (int c = tid; c < 303; c += BLK) {   // 130 + 173 B64 chunks
    const float* g;
    unsigned loff;
    if (c < 130) {
      g = gA + 2 * c;
      loff = (unsigned)(unsigned long long)(const void*)&sPred[2 * c];
    } else {
      const int cc = c - 130;
      g = gB + 2 * cc;
      loff = (unsigned)(unsigned long long)(const void*)&sPred[Hh + 2 * cc];
    }
    const unsigned long long ga = (unsigned long long)g;
    asm volatile("global_load_async_to_lds_b64 %0, %1, off"
                 :: "v"(loff), "v"(ga) : "memory");
  }
  asm volatile("s_wait_asynccnt 0" ::: "memory");
  __syncthreads();

  // ---- windowed reverse-decay accumulation into LDS (ds_add_f32) ----
  // s_t[c] = sum_{j>=t, t_j - t_t <= THR, p_j == p} exp((t_t - t_j)/TEMP) at
  // channels y_j (H part) and 260 + x_j (W part). t==0 keeps only its own base.
  const int jEnd = (t == 0) ? 1 : Tt;
  for (int j = t + tid; j < jEnd; j += BLK) {
    const float4 e = ((const float4*)target)[b * Tt + j];
    const float dt = e.x - t0;
    if (dt > WIN_THR) break;            // timestamps sorted -> safe early exit
    if ((int)e.w == p) {
      const float w = expf(dt * (-1.0f / TEMP_F));
      atomicAdd(&sS[(int)e.y], w);
      atomicAdd(&sS[Hh + (int)e.z], w);
    }
  }
  __syncthreads();

  // ---- per-segment soft-target cross entropy: lse(pred) - softmax(s).pred ----
  float lossTotal = 0.0f;
#pragma unroll
  for (int sg = 0; sg < 2; ++sg) {
    const int s0 = sg ? Hh : 0;
    const int s1 = sg ? SEGTOT : Hh;

    float mp = -3.402823e38f;
    for (int i = s0 + tid; i < s1; i += BLK) mp = fmaxf(mp, sPred[i]);
    mp = waveMax(mp);
    if (lane == 0) red[wid] = mp;
    __syncthreads();
    float m = red[0];
#pragma unroll
    for (int k = 1; k < NWAVE; ++k) m = fmaxf(m, red[k]);
    __syncthreads();

    float sE = 0.0f, sD = 0.0f, sN = 0.0f;
    for (int i = s0 + tid; i < s1; i += BLK) {
      const float pv = sPred[i];
      sE += expf(pv - m);
      const float sv = sS[i];
      const float es = (sv == 0.0f) ? 1.0f : expf(sv);  // s >= 0, small
      sD += es;
      sN += es * pv;
    }
    sE = waveSum(sE); sD = waveSum(sD); sN = waveSum(sN);
    if (lane == 0) { red[wid] = sE; red[NWAVE + wid] = sD; red[2 * NWAVE + wid] = sN; }
    __syncthreads();
    float tE = 0.0f, tD = 0.0f, tN = 0.0f;
#pragma unroll
    for (int k = 0; k < NWAVE; ++k) {
      tE += red[k]; tD += red[NWAVE + k]; tN += red[2 * NWAVE + k];
    }
    __syncthreads();
    lossTotal += (m + logf(tE)) - tN / tD;
  }

  if (tid == 0) {
    const float v = lossTotal * (1.0f / (Bb * Tt));
    if (atomicMode) atomicAdd(outbuf, v);
    else outbuf[bid] = v;
  }
}

__global__ __launch_bounds__(BLK) void tdloss_reduce(
    const float* __restrict__ partial, float* __restrict__ out, int n) {
  __shared__ float buf[BLK];
  float a = 0.0f;
  for (int i = threadIdx.x; i < n; i += BLK) a += partial[i];
  buf[threadIdx.x] = a;
  __syncthreads();
  for (int s = BLK / 2; s > 0; s >>= 1) {
    if (threadIdx.x < s) buf[threadIdx.x] += buf[threadIdx.x + s];
    __syncthreads();
  }
  if (threadIdx.x == 0) out[0] = buf[0];
}

__global__ void tdloss_zero(float* out) {
  if (threadIdx.x == 0 && blockIdx.x == 0) out[0] = 0.0f;
}

extern "C" void kernel_launch(void* const* d_in, const int* in_sizes, int n_in,
                              void* d_out, int out_size, void* d_ws, size_t ws_size,
                              hipStream_t stream) {
  (void)in_sizes; (void)n_in; (void)out_size;
  const float* pred   = (const float*)d_in[0];
  const float* target = (const float*)d_in[1];
  float* out = (float*)d_out;

  const int nBlocks = Bb * Tt;  // 65536
  if (ws_size >= (size_t)nBlocks * sizeof(float)) {
    float* partial = (float*)d_ws;
    tdloss_main<<<nBlocks, BLK, 0, stream>>>(pred, target, partial, 0);
    tdloss_reduce<<<1, BLK, 0, stream>>>(partial, out, nBlocks);
  } else {
    tdloss_zero<<<1, 32, 0, stream>>>(out);
    tdloss_main<<<nBlocks, BLK, 0, stream>>>(pred, target, out, 1);
  }
}